// PolygonGCN_26938034880619
// MI455X (gfx1250) — compile-verified
//
#include <hip/hip_runtime.h>

typedef __attribute__((ext_vector_type(2)))  float v2f;
typedef __attribute__((ext_vector_type(8)))  float v8f;

// ---------------------------------------------------------------------------
// WMMA GEMM: out[r, ocol + j] = (relu?)( A[r, 0:K] @ W[K, OUTC] + bias[j] )
// A row-major stride lda; W row-major [K, OUTC]; out stride ldo.
// One wave computes 16 rows x (16*NT) cols; block = 32x8 = 8 waves (wave32).
// OUTC is compile-time so all B addresses are immediate offsets from one
// pointer (avoids flat_load + per-element 64-bit address math).
// Preconditions: K % 4 == 0, OUTC % (16*NT) == 0, A 8-byte aligned.
// ---------------------------------------------------------------------------
template <int NT, int OUTC>
__global__ __launch_bounds__(256) void gemm_wmma_k(
    const float* __restrict__ A, int lda,
    const float* __restrict__ W,
    const float* __restrict__ bias,
    float* __restrict__ out, int ldo, int ocol,
    int nrows, int K, int relu)
{
  const int lane = threadIdx.x;              // 0..31
  const int wave = threadIdx.y;              // 0..7
  const int row0 = (blockIdx.x * 8 + wave) * 16;
  const int col0 = blockIdx.y * (16 * NT);
  if (row0 >= nrows) return;                 // wave-uniform exit, EXEC stays all-1
  const int hs  = lane >> 4;                 // lane-half select
  const int l15 = lane & 15;

  // Clamp instead of masking: clamped lanes feed rows that are never stored,
  // so no exec-mask branching is needed around the loads.
  int mrow = row0 + l15;
  if (mrow >= nrows) mrow = nrows - 1;

  // f32 A 16x4 layout: lanes 0-15 hold K={k,k+1}, lanes 16-31 hold K={k+2,k+3}
  const float* __restrict__ Ap = A + (size_t)mrow * lda + hs * 2;
  // Single B base pointer; both column tiles reached via constant offsets.
  const float* __restrict__ Wb = W + (size_t)(hs * 2) * OUTC + col0 + l15;

  v8f acc0 = {};
  v8f acc1 = {};

  #pragma unroll 4
  for (int k0 = 0; k0 < K; k0 += 4) {
    const v2f a = *(const v2f*)Ap;           // one b64 load for both K elems
    Ap += 4;
    v2f b0;
    b0.x = Wb[0];
    b0.y = Wb[OUTC];
    acc0 = __builtin_amdgcn_wmma_f32_16x16x4_f32(false, a, false, b0,
                                                 (short)0, acc0, false, false);
    if (NT == 2) {
      v2f b1;
      b1.x = Wb[16];
      b1.y = Wb[OUTC + 16];
      acc1 = __builtin_amdgcn_wmma_f32_16x16x4_f32(false, a, false, b1,
                                                   (short)0, acc1, false, false);
    }
    Wb += 4 * OUTC;
  }

  // C/D layout: VGPR i -> row row0 + hs*8 + i, col = tile_col + (lane & 15)
  const int rbase = row0 + hs * 8;
  #pragma unroll
  for (int t = 0; t < NT; ++t) {
    const v8f& acc = (t == 0) ? acc0 : acc1;
    const int ncol = col0 + t * 16 + l15;
    const float bc = bias ? bias[ncol] : 0.0f;
    #pragma unroll
    for (int i = 0; i < 8; ++i) {
      const int r = rbase + i;
      if (r < nrows) {
        float v = acc[i] + bc;
        if (relu) v = fmaxf(v, 0.0f);
        out[(size_t)r * ldo + ocol + ncol] = v;
      }
    }
  }
}

// x-projection (K=3, too small for WMMA): H[:,0:32] = relu(x @ Wx + bx)
__global__ void xproj_k(const float* __restrict__ x, const float* __restrict__ Wx,
                        const float* __restrict__ bx, float* __restrict__ H, int n)
{
  int idx = blockIdx.x * blockDim.x + threadIdx.x;
  if (idx >= n * 32) return;
  const int node = idx >> 5, c = idx & 31;
  const float x0 = x[node * 3 + 0], x1 = x[node * 3 + 1], x2 = x[node * 3 + 2];
  float v = bx[c] + x0 * Wx[0 * 32 + c] + x1 * Wx[1 * 32 + c] + x2 * Wx[2 * 32 + c];
  H[(size_t)node * 144 + c] = fmaxf(v, 0.0f);
}

__global__ void deg_init_k(float* __restrict__ deg, int n) {
  int i = blockIdx.x * blockDim.x + threadIdx.x;
  if (i < n) deg[i] = 1.0f;   // self-loop baked in
}
__global__ void deg_scatter_k(const int* __restrict__ dst, float* __restrict__ deg, int E) {
  int e = blockIdx.x * blockDim.x + threadIdx.x;
  if (e < E) atomicAdd(&deg[dst[e]], 1.0f);
}
__global__ void deg_rsqrt_k(float* __restrict__ deg, int n) {
  int i = blockIdx.x * blockDim.x + threadIdx.x;
  if (i < n) deg[i] = rsqrtf(deg[i]);
}
__global__ void zero_k(float* __restrict__ p, int n) {
  int i = blockIdx.x * blockDim.x + threadIdx.x;
  if (i < n) p[i] = 0.0f;
}

// msg scatter: agg[dst] += T[src] * dinv[src]*dinv[dst]   (4 channels / thread)
__global__ void edge_scatter_k(const int* __restrict__ src, const int* __restrict__ dst,
                               const float* __restrict__ dinv, const float* __restrict__ T,
                               float* __restrict__ agg, int E, int C)
{
  const int groups = C >> 2;
  const long long idx = (long long)blockIdx.x * blockDim.x + threadIdx.x;
  if (idx >= (long long)E * groups) return;
  const int e = (int)(idx / groups);
  const int g = (int)(idx % groups);
  const int s = src[e], d = dst[e];
  const float* tp = T + (size_t)s * C + g * 4;
  __builtin_prefetch(tp, 0, 1);
  const float norm = dinv[s] * dinv[d];
  const float4 m = *(const float4*)tp;
  float* ap = agg + (size_t)d * C + g * 4;
  atomicAdd(ap + 0, m.x * norm);
  atomicAdd(ap + 1, m.y * norm);
  atomicAdd(ap + 2, m.z * norm);
  atomicAdd(ap + 3, m.w * norm);
}

// out[n, ocol+c] = relu(agg[n,c] + T[n,c]*dinv[n]^2 + b[c])
__global__ void combine_k(const float* __restrict__ agg, const float* __restrict__ T,
                          const float* __restrict__ dinv, const float* __restrict__ bias,
                          float* __restrict__ dstbuf, int ldd, int ocol, int n, int C)
{
  int idx = blockIdx.x * blockDim.x + threadIdx.x;
  if (idx >= n * C) return;
  const int node = idx / C, c = idx % C;
  const float di = dinv[node];
  float v = agg[idx] + T[idx] * di * di + bias[c];
  dstbuf[(size_t)node * ldd + ocol + c] = fmaxf(v, 0.0f);
}

__global__ void copy_k(const float* __restrict__ s, float* __restrict__ d, int n) {
  int i = blockIdx.x * blockDim.x + threadIdx.x;
  if (i < n) d[i] = s[i];
}

extern "C" void kernel_launch(void* const* d_in, const int* in_sizes, int n_in,
                              void* d_out, int out_size, void* d_ws, size_t ws_size,
                              hipStream_t stream)
{
  const int N = in_sizes[0] / 3;
  const int E = in_sizes[6] / 2;

  // Workspace (fp32): dinv | H[N,144] | T[N,64] | AGG[N,64] | M1[N,64]
  float* ws   = (float*)d_ws;
  float* dinv = ws;
  float* H    = dinv + N;
  float* T    = H   + (size_t)N * 144;
  float* AGG  = T   + (size_t)N * 64;
  float* M1   = AGG + (size_t)N * 64;
  float* outp = (float*)d_out;

  const float* Wx  = (const float*)d_in[8];
  const float* bx  = (const float*)d_in[9];
  const float* Wf1 = (const float*)d_in[10];
  const float* bf1 = (const float*)d_in[11];
  const float* Wf2 = (const float*)d_in[12];
  const float* bf2 = (const float*)d_in[13];

  const int Kl[4] = {48, 80, 112, 144};
  const dim3 blk(32, 8);
  const int rowTiles = (N + 15) / 16;
  const unsigned gx = (unsigned)((rowTiles + 7) / 8);

  for (int gi = 0; gi < 2; ++gi) {
    const float* x  = (const float*)d_in[3 * gi + 0];
    const float* f  = (const float*)d_in[3 * gi + 1];
    const float* gv = (const float*)d_in[3 * gi + 2];
    const int*   ei = (const int*)d_in[6 + gi];
    const int* src = ei;
    const int* dst = ei + E;

    // Degrees on A+I, then dinv = rsqrt(deg)
    deg_init_k<<<(N + 255) / 256, 256, 0, stream>>>(dinv, N);
    deg_scatter_k<<<(E + 255) / 256, 256, 0, stream>>>(dst, dinv, E);
    deg_rsqrt_k<<<(N + 255) / 256, 256, 0, stream>>>(dinv, N);

    // Node MLPs -> M1 = relu(f@Wf1+bf1); H[:,32:48] = relu(M1@Wf2+bf2); H[:,0:32] = x-proj
    gemm_wmma_k<2, 64><<<dim3(gx, 2), blk, 0, stream>>>(f, 128, Wf1, bf1,
                                                        M1, 64, 0, N, 128, 1);
    gemm_wmma_k<1, 16><<<dim3(gx, 1), blk, 0, stream>>>(M1, 64, Wf2, bf2,
                                                        H, 144, 32, N, 64, 1);
    xproj_k<<<(N * 32 + 255) / 256, 256, 0, stream>>>(x, Wx, bx, H, N);

    // 4 GCN layers over the growing concat H[:, :K]
    for (int l = 0; l < 4; ++l) {
      const int K = Kl[l];
      const int C = (l == 3) ? 64 : 32;
      const float* Wl = (const float*)d_in[14 + 2 * l];
      const float* bl = (const float*)d_in[15 + 2 * l];

      if (l == 3)
        gemm_wmma_k<2, 64><<<dim3(gx, 2), blk, 0, stream>>>(H, 144, Wl, nullptr,
                                                            T, 64, 0, N, K, 0);
      else
        gemm_wmma_k<2, 32><<<dim3(gx, 1), blk, 0, stream>>>(H, 144, Wl, nullptr,
                                                            T, 32, 0, N, K, 0);

      zero_k<<<(N * C + 255) / 256, 256, 0, stream>>>(AGG, N * C);

      const long long tot = (long long)E * (C / 4);
      edge_scatter_k<<<(unsigned)((tot + 255) / 256), 256, 0, stream>>>(
          src, dst, dinv, T, AGG, E, C);

      if (l < 3)
        combine_k<<<(N * C + 255) / 256, 256, 0, stream>>>(
            AGG, T, dinv, bl, H, 144, 48 + 32 * l, N, C);
      else
        combine_k<<<(N * C + 255) / 256, 256, 0, stream>>>(
            AGG, T, dinv, bl, outp + (size_t)gi * N * 64, 64, 0, N, 64);
    }

    // Pass-through g vector
    copy_k<<<1, 128, 0, stream>>>(gv, outp + (size_t)2 * N * 64 + 128 * gi, 128);
  }
}